// RelGraphConvLayer_16484084482966
// MI455X (gfx1250) — compile-verified
//
#include <hip/hip_runtime.h>

typedef __attribute__((ext_vector_type(2))) float v2f;
typedef __attribute__((ext_vector_type(8))) float v8f;

#define NN 100000
#define NE 1600000
#define RR 8
#define BB 4
#define DD 128

// ---- CDNA5 async global->LDS copy (16B per lane), tracked by ASYNCcnt ----
__device__ __forceinline__ void async_copy16(unsigned lds_byte_addr, const void* gaddr) {
    asm volatile("global_load_async_to_lds_b128 %0, %1, off"
                 :: "v"(lds_byte_addr), "v"(gaddr)
                 : "memory");
}
__device__ __forceinline__ void wait_async0() {
    asm volatile("s_wait_asynccnt 0x0" ::: "memory");
}

// ---------------- zero workspace ----------------
__global__ void zero_f4(float4* __restrict__ p, size_t n4) {
    size_t i = (size_t)blockIdx.x * blockDim.x + threadIdx.x;
    size_t stride = (size_t)gridDim.x * blockDim.x;
    float4 z = make_float4(0.f, 0.f, 0.f, 0.f);
    for (; i < n4; i += stride) p[i] = z;
}

// ---------------- W[r,i,o] = sum_b w_comp[r,b] * bases[b,i,o] ----------------
__global__ void compute_w(const float* __restrict__ w_comp,
                          const float* __restrict__ bases,
                          float* __restrict__ W) {
    int idx = blockIdx.x * blockDim.x + threadIdx.x;   // over RR*DD*DD = 131072
    int io = idx & (DD * DD - 1);
    int r  = idx >> 14;                                // idx / (128*128)
    float acc = 0.f;
#pragma unroll
    for (int b = 0; b < BB; ++b)
        acc = fmaf(w_comp[r * BB + b], bases[(size_t)b * DD * DD + io], acc);
    W[idx] = acc;
}

// ---------------- edge scatter: one wave32 per edge ----------------
__global__ void edge_scatter(const float* __restrict__ x,
                             const int* __restrict__ src,
                             const int* __restrict__ dst,
                             const int* __restrict__ ety,
                             float* __restrict__ agg,
                             float* __restrict__ deg) {
    int lane = threadIdx.x & 31;
    int e = blockIdx.x * (blockDim.x >> 5) + (threadIdx.x >> 5);
    if (e >= NE) return;
    int s  = src[e];
    int d_ = dst[e];
    int r  = ety[e];
    int seg = d_ * RR + r;
    if (lane == 0) atomicAdd(&deg[seg], 1.0f);
    // coalesced 512B row gather (x is L2-resident: 51.2 MB < 192 MB L2)
    float4 v = ((const float4*)x)[(size_t)s * (DD / 4) + lane];
    float* base = agg + (size_t)seg * DD + lane * 4;
    atomicAdd(base + 0, v.x);
    atomicAdd(base + 1, v.y);
    atomicAdd(base + 2, v.z);
    atomicAdd(base + 3, v.w);
}

// ---------------- fused normalize + basis GEMM + self-loop + bias + ReLU ----------------
// block = 256 threads = 8 waves; block owns 16 nodes; wave w owns out cols [16w,16w+16)
// Double-buffered: relation tile rel+1 streams into LDS via async-to-LDS while
// the 32-deep WMMA chain for relation rel executes.
__global__ __launch_bounds__(256) void rgcn_gemm(
    const float* __restrict__ x,
    const float* __restrict__ agg,
    const float* __restrict__ deg,
    const float* __restrict__ W,
    const float* __restrict__ loop_w,
    const float* __restrict__ bias,
    float* __restrict__ out) {
    __shared__ float As[2][16 * DD];           // 2 x 8 KB A tiles
    __shared__ float inv_s[(RR + 1) * 16];     // 1/max(deg,1) per (rel, node); 1.0 for self-loop

    const int n0   = blockIdx.x * 16;
    const int tid  = threadIdx.x;
    const int wave = tid >> 5;
    const int lane = tid & 31;
    const int o0   = wave * 16;

    const int mh = lane & 15;                  // A: M index / B: N index
    const int kh = (lane >> 4) * 2;            // K sub-offset by lane half

    // ---- stage 1/max(deg,1) table (144 entries) ----
    if (tid < (RR + 1) * 16) {
        int m   = tid & 15;
        int rel = tid >> 4;
        float v = 1.0f;
        if (rel < RR) v = 1.0f / fmaxf(deg[(size_t)(n0 + m) * RR + rel], 1.0f);
        inv_s[tid] = v;
    }

    // ---- async-stage helper: 16x128 tile = 512 x 16B chunks, 2 per thread ----
    auto stage = [&](int rel, int buf) {
#pragma unroll
        for (int it = 0; it < 2; ++it) {
            int c  = tid + it * 256;           // chunk index 0..511
            int m  = c >> 5;                   // node within tile
            int k4 = (c & 31) * 4;             // float offset within row
            const float* g = (rel < RR)
                ? agg + ((size_t)(n0 + m) * RR + rel) * DD + k4
                : x + (size_t)(n0 + m) * DD + k4;
            unsigned lds = (unsigned)(size_t)(&As[buf][m * DD + k4]);
            async_copy16(lds, g);
        }
    };

    stage(0, 0);                               // prefetch relation 0
    wait_async0();
    __syncthreads();

    v8f c = {};

    for (int rel = 0; rel <= RR; ++rel) {
        const int cur = rel & 1;
        if (rel < RR) stage(rel + 1, cur ^ 1); // overlap next tile with compute

        const float* Bm = (rel < RR) ? (W + (size_t)rel * DD * DD) : loop_w;
        const float invd = inv_s[rel * 16 + mh];
        const float* Arow = &As[cur][mh * DD];

#pragma unroll 8
        for (int k0 = 0; k0 < DD; k0 += 4) {
            v2f a, b;
            // A 16x4 f32: lanes0-15 M=0..15 K={k0,k0+1}; lanes16-31 K={k0+2,k0+3}
            a.x = Arow[k0 + kh] * invd;
            a.y = Arow[k0 + kh + 1] * invd;
            // B 4x16 f32: lane = N, VGPR+lane-half = K  (row-major W: [k][o])
            b.x = Bm[(size_t)(k0 + kh) * DD + o0 + mh];
            b.y = Bm[(size_t)(k0 + kh + 1) * DD + o0 + mh];
            c = __builtin_amdgcn_wmma_f32_16x16x4_f32(
                    false, a, false, b, (short)0, c, false, false);
        }

        wait_async0();                         // next tile landed in LDS
        __syncthreads();                       // all waves done reading/writing
    }

    // D 16x16 f32: VGPR j -> M=j (lanes 0-15) / M=j+8 (lanes 16-31); N = lane&15
    const int mbase = (lane >> 4) * 8;
    const float bv = bias[o0 + mh];
#pragma unroll
    for (int j = 0; j < 8; ++j) {
        int m = mbase + j;
        float hv = c[j] + bv;
        out[(size_t)(n0 + m) * DD + o0 + mh] = fmaxf(hv, 0.0f);
    }
}

extern "C" void kernel_launch(void* const* d_in, const int* in_sizes, int n_in,
                              void* d_out, int out_size, void* d_ws, size_t ws_size,
                              hipStream_t stream) {
    const float* x       = (const float*)d_in[0];
    const float* w_comp  = (const float*)d_in[1];
    const float* bases   = (const float*)d_in[2];
    const float* loop_w  = (const float*)d_in[3];
    const float* h_bias  = (const float*)d_in[4];
    const int*   src     = (const int*)d_in[5];
    const int*   dst     = (const int*)d_in[6];
    const int*   ety     = (const int*)d_in[7];
    float* out = (float*)d_out;

    // workspace layout: W | deg | agg   (deg+agg zeroed contiguously)
    float* W   = (float*)d_ws;                          // RR*DD*DD        = 0.5 MB
    float* deg = W + (size_t)RR * DD * DD;              // NN*RR           = 3.2 MB
    float* agg = deg + (size_t)NN * RR;                 // NN*RR*DD        = 409.6 MB

    size_t zero_elems = (size_t)NN * RR * (DD + 1);     // deg + agg contiguous
    size_t n4 = zero_elems / 4;
    int zgrid = (int)((n4 + 255) / 256);
    if (zgrid > 65535 * 8) zgrid = 65535 * 8;
    zero_f4<<<zgrid, 256, 0, stream>>>((float4*)deg, n4);

    compute_w<<<(RR * DD * DD) / 256, 256, 0, stream>>>(w_comp, bases, W);

    edge_scatter<<<(NE + 7) / 8, 256, 0, stream>>>(x, src, dst, ety, agg, deg);

    rgcn_gemm<<<NN / 16, 256, 0, stream>>>(x, agg, deg, W, loop_w, h_bias, out);
}